// ProteinGCN_69002944577888
// MI455X (gfx1250) — compile-verified
//
#include <hip/hip_runtime.h>
#include <hip/hip_bf16.h>
#include <stdint.h>
#include <stddef.h>

typedef __attribute__((ext_vector_type(2))) float v2f;
typedef __attribute__((ext_vector_type(8))) float v8f;

#define N_NODES  20000
#define N_EDGES  160000
#define N_GRAPHS 32
#define F_IN     64
#define F_CAT    1920
#define F_OUT    6630

// ---------------------------------------------------------------- utilities
__global__ void fill_kernel(float* __restrict__ p, float v, size_t n) {
    size_t i = (size_t)blockIdx.x * blockDim.x + threadIdx.x;
    size_t stride = (size_t)gridDim.x * blockDim.x;
    for (; i < n; i += stride) p[i] = v;
}

// deg[dst] += w  (deg pre-filled with 1.0 for the self-loop weight)
__global__ void deg_accum_kernel(const int* __restrict__ ei,
                                 const float* __restrict__ w,
                                 float* __restrict__ deg, int E) {
    int e = blockIdx.x * blockDim.x + threadIdx.x;
    if (e < E) atomicAdd(&deg[ei[E + e]], w[e]);
}

// in-place deg -> 1/sqrt(deg)  (deg >= 1 always, so no zero guard needed)
__global__ void rsqrt_kernel(float* __restrict__ d, int n) {
    int i = blockIdx.x * blockDim.x + threadIdx.x;
    if (i < n) d[i] = rsqrtf(d[i]);
}

// norm[e] = dinv[src]*w*dinv[dst] for e<E ; norm[E+i] = dinv[i]^2 (self-loop)
__global__ void norm_kernel(const int* __restrict__ ei,
                            const float* __restrict__ w,
                            const float* __restrict__ dinv,
                            float* __restrict__ norm, int E, int Nn) {
    int idx = blockIdx.x * blockDim.x + threadIdx.x;
    if (idx < E) {
        norm[idx] = dinv[ei[idx]] * w[idx] * dinv[ei[E + idx]];
    } else if (idx < E + Nn) {
        float d = dinv[idx - E];
        norm[idx] = d * d;
    }
}

// ------------------------------------------------------- WMMA f32 GEMM core
// C[M,N] = A[M,K](lda) @ B[K,N](ldb) (+bias).
// Register-blocked: each wave owns a 16x64 tile = 4 accumulators; the A
// fragment is loaded once per 4-K step and reused by 4 independent
// V_WMMA_F32_16X16X4_F32 chains (hides the WMMA->WMMA RAW hazard).
// A frag (16x4): lane<16 holds K=k,k+1 of row=lane; lane>=16 holds K=k+2,k+3.
// B frag (4x16): v0 = rows k (lanes 0-15) / k+2 (lanes 16-31), v1 = k+1 / k+3.
// D layout: acc[v] -> row = m0 + (lane>>4)*8 + v, col = ntile*16 + (lane&15).
__global__ __launch_bounds__(128) void gemm_wmma_f32(
    const float* __restrict__ A, int lda,
    const float* __restrict__ B, int ldb,
    float* __restrict__ C, int ldc,
    int M, int N, int K, const float* __restrict__ bias)
{
    const int wave  = threadIdx.x >> 5;
    const int lane  = threadIdx.x & 31;
    const int row16 = lane & 15;
    const int half  = lane >> 4;          // 0 or 1
    const int m0 = blockIdx.x * 16;
    const int n0 = (blockIdx.y * 4 + wave) * 64;   // wave covers 64 columns
    if (n0 >= N) return;                  // wave-uniform exit (EXEC stays full)

    int  col[4];
    bool colok[4];
    const float* bcol[4];
    #pragma unroll
    for (int t = 0; t < 4; ++t) {
        col[t]   = n0 + t * 16 + row16;
        colok[t] = col[t] < N;
        bcol[t]  = B + (colok[t] ? col[t] : (N - 1));  // clamp: stay in-bounds
    }
    const float* arow = A + (size_t)(m0 + row16) * lda + 2 * half;

    v8f acc0 = {}, acc1 = {}, acc2 = {}, acc3 = {};
    for (int k = 0; k < K; k += 4) {
        __builtin_prefetch(arow + k + 16, 0, 0);       // global_prefetch_b8
        v2f a = *(const v2f*)(arow + k);               // K = k+2h, k+2h+1
        const size_t kk = (size_t)(k + 2 * half);
        v2f b0, b1, b2, b3;
        b0.x = bcol[0][kk * ldb];  b0.y = bcol[0][(kk + 1) * ldb];
        b1.x = bcol[1][kk * ldb];  b1.y = bcol[1][(kk + 1) * ldb];
        b2.x = bcol[2][kk * ldb];  b2.y = bcol[2][(kk + 1) * ldb];
        b3.x = bcol[3][kk * ldb];  b3.y = bcol[3][(kk + 1) * ldb];
        acc0 = __builtin_amdgcn_wmma_f32_16x16x4_f32(
                   false, a, false, b0, (short)0, acc0, false, false);
        acc1 = __builtin_amdgcn_wmma_f32_16x16x4_f32(
                   false, a, false, b1, (short)0, acc1, false, false);
        acc2 = __builtin_amdgcn_wmma_f32_16x16x4_f32(
                   false, a, false, b2, (short)0, acc2, false, false);
        acc3 = __builtin_amdgcn_wmma_f32_16x16x4_f32(
                   false, a, false, b3, (short)0, acc3, false, false);
    }

    const int rbase = m0 + half * 8;      // M is a multiple of 16 in all calls
    v8f accs[4] = { acc0, acc1, acc2, acc3 };
    #pragma unroll
    for (int t = 0; t < 4; ++t) {
        if (!colok[t]) continue;
        float bv = bias ? bias[col[t]] : 0.0f;
        #pragma unroll
        for (int v = 0; v < 8; ++v)
            C[(size_t)(rbase + v) * ldc + col[t]] = accs[t][v] + bv;
    }
}

// ----------------------------------------------------- message scatter (agg)
// agg[dst, :] += norm[e] * h[src, :]   for e in [0, E+Nn)  (E+i = self-loop i)
__global__ void scatter_kernel(const int* __restrict__ ei,
                               const float* __restrict__ norm,
                               const float* __restrict__ h,
                               float* __restrict__ agg,
                               int E, int Nn, int C) {
    int e = blockIdx.x;
    float nrm = norm[e];
    int s, d;
    if (e < E) { s = ei[e]; d = ei[E + e]; }
    else       { s = e - E; d = s; }
    const float* hs = h + (size_t)s * C;
    float* ad = agg + (size_t)d * C;
    for (int c = threadIdx.x; c < C; c += blockDim.x)
        atomicAdd(&ad[c], nrm * hs[c]);
}

// xc[i, off+c] = relu(agg[i,c] + b[c])   (writes straight into JK-concat buf)
__global__ void bias_relu_cat_kernel(const float* __restrict__ agg,
                                     const float* __restrict__ b,
                                     float* __restrict__ xc,
                                     int Nn, int Cw, int off) {
    size_t idx = (size_t)blockIdx.x * blockDim.x + threadIdx.x;
    size_t tot = (size_t)Nn * Cw;
    if (idx >= tot) return;
    int i = (int)(idx / Cw);
    int c = (int)(idx % Cw);
    float v = agg[idx] + b[c];
    xc[(size_t)i * F_CAT + off + c] = v > 0.0f ? v : 0.0f;
}

// global mean pool; batch is sorted so each graph is a contiguous node range
__global__ void pool_kernel(const float* __restrict__ xc,
                            const int* __restrict__ batch,
                            float* __restrict__ pooled, int Nn) {
    int g = blockIdx.x;
    int c = blockIdx.y * blockDim.x + threadIdx.x;     // < F_CAT
    int lo = 0, hi = Nn;
    while (lo < hi) { int m = (lo + hi) >> 1; if (batch[m] <  g) lo = m + 1; else hi = m; }
    int start = lo;
    lo = start; hi = Nn;
    while (lo < hi) { int m = (lo + hi) >> 1; if (batch[m] <= g) lo = m + 1; else hi = m; }
    int end = lo;
    float sum = 0.0f;
    for (int i = start; i < end; ++i) sum += xc[(size_t)i * F_CAT + c];
    float cnt = (float)(end - start);
    pooled[(size_t)g * F_CAT + c] = sum / fmaxf(cnt, 1.0f);
}

// --------------------------------------------------------------------- host
extern "C" void kernel_launch(void* const* d_in, const int* in_sizes, int n_in,
                              void* d_out, int out_size, void* d_ws, size_t ws_size,
                              hipStream_t stream) {
    const float* x     = (const float*)d_in[0];
    const int*   ei    = (const int*)  d_in[1];   // [2, E] flat: src row then dst row
    const float* ea    = (const float*)d_in[2];
    const int*   batch = (const int*)  d_in[3];
    const float* W1 = (const float*)d_in[4];  const float* b1 = (const float*)d_in[5];
    const float* W2 = (const float*)d_in[6];  const float* b2 = (const float*)d_in[7];
    const float* W3 = (const float*)d_in[8];  const float* b3 = (const float*)d_in[9];
    const float* W4 = (const float*)d_in[10]; const float* b4 = (const float*)d_in[11];
    const float* Wlin = (const float*)d_in[12];
    const float* blin = (const float*)d_in[13];
    float* out = (float*)d_out;

    const int Nn = in_sizes[0] / F_IN;      // 20000
    const int E  = in_sizes[1] / 2;         // 160000

    // workspace carving (256B aligned)
    char* base = (char*)d_ws;
    auto carve = [&](size_t bytes) -> char* {
        char* p = base;
        base += (bytes + 255) & ~(size_t)255;
        return p;
    };
    float* deg    = (float*)carve((size_t)Nn * 4);                 // -> dinv in place
    float* norm   = (float*)carve((size_t)(E + Nn) * 4);
    float* h      = (float*)carve((size_t)Nn * 1024 * 4);          // GEMM out
    float* agg    = (float*)carve((size_t)Nn * 1024 * 4);          // scatter target
    float* xc     = (float*)carve((size_t)Nn * F_CAT * 4);         // JK concat
    float* pooled = (float*)carve((size_t)N_GRAPHS * F_CAT * 4);

    // 1) symmetric GCN normalization
    fill_kernel<<<(Nn + 255) / 256, 256, 0, stream>>>(deg, 1.0f, (size_t)Nn);
    deg_accum_kernel<<<(E + 255) / 256, 256, 0, stream>>>(ei, ea, deg, E);
    rsqrt_kernel<<<(Nn + 255) / 256, 256, 0, stream>>>(deg, Nn);
    norm_kernel<<<(E + Nn + 255) / 256, 256, 0, stream>>>(ei, ea, deg, norm, E, Nn);

    // 2) four GCN layers
    struct Layer { const float* A; int lda; const float* W; const float* b; int K; int C; int off; };
    Layer L[4] = {
        { x,        F_IN,  W1, b1, F_IN, 128,  0   },
        { xc + 0,   F_CAT, W2, b2, 128,  256,  128 },
        { xc + 128, F_CAT, W3, b3, 256,  512,  384 },
        { xc + 384, F_CAT, W4, b4, 512,  1024, 896 },
    };
    for (int l = 0; l < 4; ++l) {
        const int C = L[l].C;
        const int ntiles64 = (C + 63) / 64;            // 16x64 tile per wave
        dim3 ggrid(Nn / 16, (ntiles64 + 3) / 4);
        gemm_wmma_f32<<<ggrid, 128, 0, stream>>>(
            L[l].A, L[l].lda, L[l].W, C, h, C, Nn, C, L[l].K, nullptr);

        size_t tot = (size_t)Nn * C;
        fill_kernel<<<4096, 256, 0, stream>>>(agg, 0.0f, tot);
        scatter_kernel<<<E + Nn, 128, 0, stream>>>(ei, norm, h, agg, E, Nn, C);
        bias_relu_cat_kernel<<<(unsigned)((tot + 255) / 256), 256, 0, stream>>>(
            agg, L[l].b, xc, Nn, C, L[l].off);
    }

    // 3) global mean pool  (F_CAT = 15 * 128)
    pool_kernel<<<dim3(N_GRAPHS, F_CAT / 128), 128, 0, stream>>>(xc, batch, pooled, Nn);

    // 4) final linear: [32,1920] @ [1920,6630] + bias  (N not /64 -> guarded)
    const int ntiles64 = (F_OUT + 63) / 64;            // 104
    dim3 fgrid(N_GRAPHS / 16, (ntiles64 + 3) / 4);     // (2, 26)
    gemm_wmma_f32<<<fgrid, 128, 0, stream>>>(
        pooled, F_CAT, Wlin, F_OUT, out, F_OUT, N_GRAPHS, F_OUT, F_CAT, blin);
}